// DifferentialQuadraticSplineStack_17660905521235
// MI455X (gfx1250) — compile-verified
//
#include <hip/hip_runtime.h>
#include <math.h>

#define R_ 16
#define G_ 5000
#define SUMB 224
#define SUMW 221

typedef __attribute__((ext_vector_type(2))) float v2f;
typedef __attribute__((ext_vector_type(8))) float v8f;

// ---------------------------------------------------------------------------
// Kernel 1: per-gene softmax of unnormalized widths (shared across reflatents)
// One wave32 per gene.
// ---------------------------------------------------------------------------
__global__ void softmax_w_kernel(const float* __restrict__ uw,
                                 float* __restrict__ wnorm,
                                 int n, int w_off) {
  int wave = (blockIdx.x * blockDim.x + threadIdx.x) >> 5;
  int lane = threadIdx.x & 31;
  if (wave >= G_) return;
  int m = n - 1;
  const float* src = uw + (size_t)wave * SUMW + w_off;
  float x[4];
  float mx = -INFINITY;
  #pragma unroll
  for (int u = 0; u < 4; ++u) {
    int j = lane + 32 * u;
    x[u] = (j < m) ? src[j] : -INFINITY;
    mx = fmaxf(mx, x[u]);
  }
  #pragma unroll
  for (int d = 16; d >= 1; d >>= 1) mx = fmaxf(mx, __shfl_xor(mx, d, 32));
  float e[4];
  float s = 0.0f;
  #pragma unroll
  for (int u = 0; u < 4; ++u) {
    int j = lane + 32 * u;
    e[u] = (j < m) ? __expf(x[u] - mx) : 0.0f;
    s += e[u];
  }
  #pragma unroll
  for (int d = 16; d >= 1; d >>= 1) s += __shfl_xor(s, d, 32);
  float inv = 1.0f / s;
  float* dst = wnorm + (size_t)wave * m;
  #pragma unroll
  for (int u = 0; u < 4; ++u) {
    int j = lane + 32 * u;
    if (j < m) dst[j] = e[u] * inv;
  }
}

// ---------------------------------------------------------------------------
// Kernel 2: per-(r,g) table build. One wave32 per (r,g).
// - Parameter rows (uh, dh, wnorm) staged memory->LDS with
//   GLOBAL_LOAD_ASYNC_TO_LDS_B128 (ASYNCcnt), fenced by s_wait_asynccnt.
// - Dual exclusive prefix scan (widths -> locations, trapezoid -> local CDF)
//   via chained V_WMMA_F32_16X16X4_F32 with a strictly-lower-triangular
//   ones A-matrix: columns 0-7 of B = width chunks, columns 8-15 = cdf chunks.
// ---------------------------------------------------------------------------
__global__ void build_kernel(const float* __restrict__ uh,
                             const float* __restrict__ dh,
                             const float* __restrict__ wnorm,
                             const float* __restrict__ gs,
                             const float* __restrict__ gsL,
                             float* __restrict__ loc,
                             float* __restrict__ hgt,
                             float* __restrict__ cdf,
                             float* __restrict__ sbuf,
                             int n, int hb_off, int first) {
  __shared__ float sw[8][128];    // width elements (padded with zeros)
  __shared__ float st[8][128];    // trapezoid elements (padded with zeros)
  __shared__ float suh[8][128];   // async-staged uh row
  __shared__ float sdh[8][128];   // async-staged dh row
  __shared__ float swn[8][128];   // async-staged wnorm row

  int wib  = threadIdx.x >> 5;
  int lane = threadIdx.x & 31;
  int rg   = blockIdx.x * 8 + wib;          // grid is exactly R*G/8 blocks
  int r    = rg / G_;
  int g    = rg - r * G_;
  int j0   = lane * 4;                      // lane owns elements j0..j0+3

  // --- async staging: 3x GLOBAL_LOAD_ASYNC_TO_LDS_B128 per wave -----------
  {
    const float* uhp = uh + (size_t)g  * SUMB + hb_off;
    const float* dhp = dh + (size_t)rg * SUMB + hb_off;
    const float* wnp = wnorm + (size_t)g * (n - 1);
    unsigned int l_uh = (unsigned int)(uintptr_t)&suh[wib][j0];
    unsigned int l_dh = (unsigned int)(uintptr_t)&sdh[wib][j0];
    unsigned int l_wn = (unsigned int)(uintptr_t)&swn[wib][j0];
    if (j0 < n) {
      unsigned long long g_uh = (unsigned long long)(uintptr_t)(uhp + j0);
      unsigned long long g_dh = (unsigned long long)(uintptr_t)(dhp + j0);
      asm volatile("global_load_async_to_lds_b128 %0, %1, off"
                   :: "v"(l_uh), "v"(g_uh) : "memory");
      asm volatile("global_load_async_to_lds_b128 %0, %1, off"
                   :: "v"(l_dh), "v"(g_dh) : "memory");
    }
    if (j0 < n - 1) {
      unsigned long long g_wn = (unsigned long long)(uintptr_t)(wnp + j0);
      asm volatile("global_load_async_to_lds_b128 %0, %1, off"
                   :: "v"(l_wn), "v"(g_wn) : "memory");
    }
  }

  float gsv   = first ? (1.0f / G_) : gs[rg];
  float Lbase = first ? (g * (1.0f / G_)) : gsL[rg];
  size_t base = (size_t)rg * n;

  // zero scan staging while async copies fly
  #pragma unroll
  for (int u = 0; u < 4; ++u) {
    sw[wib][j0 + u] = 0.0f;
    st[wib][j0 + u] = 0.0f;
  }

  asm volatile("s_wait_asynccnt 0x0" ::: "memory");

  if (j0 < n) {
    float h[5];
    #pragma unroll
    for (int u = 0; u < 5; ++u) {
      int j = j0 + u;
      h[u] = (j < n) ? __expf(suh[wib][j] + sdh[wib][j]) : 0.0f;
    }
    #pragma unroll
    for (int u = 0; u < 4; ++u) {
      int j = j0 + u;
      if (j < n) hgt[base + j] = h[u];        // unnormalized; finalize divides
    }
    #pragma unroll
    for (int u = 0; u < 4; ++u) {
      int j = j0 + u;
      if (j < n - 1) {
        float w = swn[wib][j] * gsv;
        sw[wib][j] = w;
        st[wib][j] = 0.5f * (h[u] + h[u + 1]) * w;
      }
    }
  }
  __syncthreads();

  // WMMA scan: D = Lstrict(16x16) x M(16x16), accumulated over K in 4 steps.
  int c = lane & 15;                          // column handled by this lane
  const float* colbuf = (c < 8) ? &sw[wib][16 * c] : &st[wib][16 * (c - 8)];
  int i  = lane & 15;                         // A-matrix row
  int kb = (lane < 16) ? 0 : 2;               // K sub-offset per VGPR layout

  v8f acc = {0.f, 0.f, 0.f, 0.f, 0.f, 0.f, 0.f, 0.f};
  #pragma unroll
  for (int q = 0; q < 4; ++q) {
    int k0 = 4 * q + kb;
    int k1 = k0 + 1;
    v2f a, b;
    a.x = (k0 < i) ? 1.0f : 0.0f;             // strictly lower triangular ones
    a.y = (k1 < i) ? 1.0f : 0.0f;
    b.x = colbuf[k0];
    b.y = colbuf[k1];
    acc = __builtin_amdgcn_wmma_f32_16x16x4_f32(
        false, a, false, b, (short)0, acc, false, false);
  }

  // Chunk totals and per-group (w-group cols 0-7, t-group cols 8-15)
  // exclusive chunk offsets.
  float T = 0.0f;
  if (lane < 16) {
    #pragma unroll
    for (int k = 0; k < 16; ++k) T += colbuf[k];
  }
  float inc = T;
  #pragma unroll
  for (int d = 1; d < 8; d <<= 1) {
    float v = __shfl_up(inc, d, 32);
    if ((lane & 7) >= d) inc += v;
  }
  float Toff = inc - T;                        // exclusive offset (lanes 0-15)
  Toff = __shfl(Toff, lane & 15, 32);          // broadcast to partner lanes

  // Emit: lane l<16 holds rows 0-7 of column c; lane l>=16 rows 8-15.
  int cc = (c < 8) ? c : (c - 8);
  #pragma unroll
  for (int v = 0; v < 8; ++v) {
    int row = (lane < 16) ? v : (v + 8);
    int e   = 16 * cc + row;                   // element index within gene
    float val = acc[v] + Toff;
    if (e < n) {
      if (c < 8) loc[base + e] = val + Lbase;  // exclusive scan of widths
      else       cdf[base + e] = val;          // exclusive scan of trapezoids
    }
  }
  // gene mass s = exclusive cdf scan at element n-1 (last trapezoid is pad 0)
  int ct = (n - 1) >> 4;
  if (lane == 24 + ct) sbuf[rg] = acc[7] + Toff;
}

// ---------------------------------------------------------------------------
// Kernel 3: per-reflatent exclusive scan of gene masses -> P, area A,
// and next layer's genespacing / location base.
// ---------------------------------------------------------------------------
__global__ void scan_kernel(const float* __restrict__ sbuf,
                            float* __restrict__ Pex,
                            float* __restrict__ Arow,
                            float* __restrict__ gsn,
                            float* __restrict__ gsLn) {
  __shared__ float sm[256];
  __shared__ float carry_sh;
  int r = blockIdx.x;
  int t = threadIdx.x;
  const float* s = sbuf + (size_t)r * G_;
  float* P = Pex + (size_t)r * G_;
  if (t == 0) carry_sh = 0.0f;
  __syncthreads();
  for (int c0 = 0; c0 < G_; c0 += 256) {
    float carry = carry_sh;
    int g = c0 + t;
    float v = (g < G_) ? s[g] : 0.0f;
    sm[t] = v;
    __syncthreads();
    float x = v;
    for (int d = 1; d < 256; d <<= 1) {
      float y = (t >= d) ? sm[t - d] : 0.0f;
      __syncthreads();
      x += y;
      sm[t] = x;
      __syncthreads();
    }
    if (g < G_) P[g] = carry + x - v;          // exclusive prefix
    __syncthreads();
    if (t == 255) carry_sh = carry + x;
    __syncthreads();
  }
  float A = carry_sh;
  if (t == 0) Arow[r] = A;
  float invA = 1.0f / A;
  for (int g = t; g < G_; g += 256) {
    gsn [(size_t)r * G_ + g] = s[g] * invA;    // next-layer genespacing
    gsLn[(size_t)r * G_ + g] = P[g] * invA;    // next-layer location base
  }
}

// ---------------------------------------------------------------------------
// Kernel 4: normalize heights by area, add gene prefix to CDF.
// ---------------------------------------------------------------------------
__global__ void finalize_kernel(float* __restrict__ hgt,
                                float* __restrict__ cdf,
                                const float* __restrict__ Pex,
                                const float* __restrict__ Arow,
                                int n, size_t tot) {
  size_t idx = (size_t)blockIdx.x * blockDim.x + threadIdx.x;
  if (idx >= tot) return;
  size_t rg = idx / n;
  int r = (int)(rg / G_);
  float invA = 1.0f / Arow[r];
  hgt[idx] *= invA;
  cdf[idx] = (Pex[rg] + cdf[idx]) * invA;
}

// ---------------------------------------------------------------------------
// Kernel 5: transform one layer (binary search + quadratic spline).
// Tables for the running layer are mostly/entirely L2-resident (123/61/31 MB
// vs 192 MB L2), which is what makes the random gathers cheap on MI455X.
// ---------------------------------------------------------------------------
template <int N, int LOG2N>
__global__ void transform_kernel(const float* __restrict__ loc,
                                 const float* __restrict__ hgt,
                                 const float* __restrict__ cdf,
                                 const float* __restrict__ xin,
                                 const int* __restrict__ gene,
                                 const int* __restrict__ refl,
                                 float* __restrict__ xout,
                                 float* __restrict__ lad,
                                 int ncuts, int first) {
  int i = blockIdx.x * blockDim.x + threadIdx.x;
  if (i >= ncuts) return;
  float x = xin[i];
  int g = gene[i];
  int r = refl[i];
  size_t base = ((size_t)r * G_ + g) * N;
  const float* row = loc + base;
  __builtin_prefetch(&row[0], 0, 1);           // global_prefetch_b8
  __builtin_prefetch(&row[N / 2], 0, 1);

  // branchless binary search (matches reference searchsorted, right=False)
  int lo = 0, hi = N;
  #pragma unroll
  for (int it = 0; it < LOG2N + 1; ++it) {
    int mid = (lo + hi) >> 1;
    int mc = (mid < N - 1) ? mid : (N - 1);
    float tm = row[mc];
    bool right = tm < x;
    bool active = lo < hi;
    lo = (active && right) ? (mid + 1) : lo;
    hi = (active && !right) ? mid : hi;
  }
  int j = lo - 1;
  j = (j < 0) ? 0 : j;
  j = (j > N - 2) ? (N - 2) : j;

  float bl = row[j];
  float bw = row[j + 1] - bl;
  float bc = cdf[base + j];
  float hl = hgt[base + j];
  float hr = hgt[base + j + 1];
  float alpha = (x - bl) / bw;
  float y = bc + alpha * hl * bw + 0.5f * alpha * alpha * (hr - hl) * bw;
  float l = __logf(hl + alpha * (hr - hl));
  xout[i] = y;
  lad[i] = first ? l : (lad[i] + l);
}

// ---------------------------------------------------------------------------
extern "C" void kernel_launch(void* const* d_in, const int* in_sizes, int n_in,
                              void* d_out, int out_size, void* d_ws, size_t ws_size,
                              hipStream_t stream) {
  (void)n_in; (void)out_size; (void)ws_size;
  const float* cutpos = (const float*)d_in[0];
  const int*   gene   = (const int*)d_in[2];
  const int*   refl   = (const int*)d_in[3];
  const float* dh     = (const float*)d_in[4];
  const float* uh     = (const float*)d_in[5];
  const float* uw     = (const float*)d_in[6];
  int N = in_sizes[0];

  float* out = (float*)d_out;
  float* lad = out + N;

  // workspace carve (single-layer tables reused across layers)
  const size_t cap = (size_t)R_ * G_ * 128;
  float* ws   = (float*)d_ws;
  float* loc  = ws; ws += cap;
  float* hgt  = ws; ws += cap;
  float* cdf  = ws; ws += cap;
  float* sbuf = ws; ws += (size_t)R_ * G_;
  float* Pex  = ws; ws += (size_t)R_ * G_;
  float* Arow = ws; ws += 64;
  float* gsn  = ws; ws += (size_t)R_ * G_;
  float* gsLn = ws; ws += (size_t)R_ * G_;
  float* wnrm = ws; ws += (size_t)G_ * 127;

  const int nb[3] = {128, 64, 32};
  int h_off = 0, w_off = 0;
  for (int k = 0; k < 3; ++k) {
    int n = nb[k];
    softmax_w_kernel<<<(G_ * 32 + 255) / 256, 256, 0, stream>>>(uw, wnrm, n, w_off);
    build_kernel<<<(R_ * G_) / 8, 256, 0, stream>>>(uh, dh, wnrm, gsn, gsLn,
                                                    loc, hgt, cdf, sbuf,
                                                    n, h_off, (k == 0) ? 1 : 0);
    scan_kernel<<<R_, 256, 0, stream>>>(sbuf, Pex, Arow, gsn, gsLn);
    size_t tot = (size_t)R_ * G_ * n;
    finalize_kernel<<<(unsigned)((tot + 255) / 256), 256, 0, stream>>>(hgt, cdf, Pex, Arow, n, tot);

    const float* xin = (k == 0) ? cutpos : out;
    int grid = (N + 255) / 256;
    if (n == 128)
      transform_kernel<128, 7><<<grid, 256, 0, stream>>>(loc, hgt, cdf, xin, gene, refl, out, lad, N, k == 0);
    else if (n == 64)
      transform_kernel<64, 6><<<grid, 256, 0, stream>>>(loc, hgt, cdf, xin, gene, refl, out, lad, N, k == 0);
    else
      transform_kernel<32, 5><<<grid, 256, 0, stream>>>(loc, hgt, cdf, xin, gene, refl, out, lad, N, k == 0);

    h_off += n;
    w_off += n - 1;
  }
}